// RVBProj_angular_8856222565068
// MI455X (gfx1250) — compile-verified
//
#include <hip/hip_runtime.h>
#include <hip/hip_bf16.h>
#include <math.h>

typedef __attribute__((ext_vector_type(2))) float v2f;
typedef __attribute__((ext_vector_type(8))) float v8f;

// RVB basis state indices; their 2-bit fields ARE the per-triplet digits.
__device__ __constant__ unsigned short RVB_IDX_c[32] = {
    4005, 4090, 5657, 5702, 10043, 10084, 16007, 16088,
    18713, 18758, 20645, 20730, 24967, 25048, 30779, 30820,
    36241, 36302, 37933, 38002, 42255, 42320, 48307, 48364,
    52013, 52082, 53905, 53966, 58291, 58348, 64015, 64080};

__device__ __forceinline__ float2 clogf2(float re, float im) {
  return make_float2(0.5f * logf(re * re + im * im), atan2f(im, re));
}

// LM[i][j]: i = query digit (row of M), j = RVB digit (col of M), complex log.
__device__ __forceinline__ float2 lmsel(int i, int j, float2 lA, float2 lB,
                                        float2 lC, float2 lD, float2 lE) {
  float2 r;
  r = (i == j) ? lD : lE;          // i>0, j>0 cases
  r = (i == 0) ? lC : r;           // first row (j>0)
  if (j == 0) r = (i == 0) ? lA : lB;
  return r;
}

__global__ __launch_bounds__(32) void rvb_logpsi_kernel(
    const int* __restrict__ x_in, const float* __restrict__ r1p,
    const float* __restrict__ r2p, const float* __restrict__ t1p,
    const float* __restrict__ t2p, float* __restrict__ out) {
  const int lane = threadIdx.x;  // wave32
  const int half = lane >> 4;    // 0: K/rows 0..; 1: K/rows +
  const int ln = lane & 15;
  const int qbase = blockIdx.x * 16;

  // ---- scalar log-M table (identical in all lanes) ----
  const float r1 = r1p[0], r2 = r2p[0], t1 = t1p[0], t2 = t2p[0];
  const float c1 = cosf(r1), c2 = cosf(r2);
  const float sr1 = sinf(r1), sr2 = sinf(r2);
  const float s1re = sr1 * cosf(t1), s1im = sr1 * sinf(t1);
  const float s2re = sr2 * cosf(t2), s2im = sr2 * sinf(t2);
  const float c1_2 = c1 * c1, c1_3 = c1_2 * c1;
  const float c2_2 = c2 * c2, c2_3 = c2_2 * c2;
  const float ssre = s1re * s2re - s1im * s2im;   // s1*s2
  const float ssim = s1re * s2im + s1im * s2re;
  const float2 lA = clogf2(c1_3 * c2_3, 0.0f);
  const float2 lB = clogf2(c1_3 * c2_2 * s2re, c1_3 * c2_2 * s2im);
  const float2 lC = clogf2(c1_2 * c2_3 * s1re, c1_2 * c2_3 * s1im);
  const float2 lD = clogf2(c1_2 * c2_2 * (c1 * c2 + ssre), c1_2 * c2_2 * ssim);
  const float2 lE = clogf2(c1_2 * c2_2 * ssre, c1_2 * c2_2 * ssim);

  // ---- digits of my query row (row M = ln, both half-waves need same 16) ----
  const int* xr = x_in + (size_t)(qbase + ln) * 24;
  int dig[8];
#pragma unroll
  for (int t = 0; t < 8; ++t) {
    int b0 = xr[3 * t] > 0, b1 = xr[3 * t + 1] > 0, b2 = xr[3 * t + 2] > 0;
    int p = (b0 << 2) | (b1 << 1) | b2;  // in {0,1,2,4}
    dig[t] = (p == 4) ? 3 : p;
  }

  // RVB digit sources for my B-column (col N = ln; tile0: r=ln, tile1: r=ln+16)
  const unsigned rv0 = RVB_IDX_c[ln];
  const unsigned rv1 = RVB_IDX_c[ln + 16];
  const int d0 = half * 2, d1 = d0 + 1;  // local-K -> digit hypothesis

  // ---- GEMM: S = onehot(X) * LM  via 8 chained K=4 f32 WMMAs per tile/part
  v8f cr0 = {}, ci0 = {}, cr1 = {}, ci1 = {};
#pragma unroll
  for (int t = 0; t < 8; ++t) {
    v2f a;
    a.x = (dig[t] == d0) ? 1.0f : 0.0f;
    a.y = (dig[t] == d1) ? 1.0f : 0.0f;
    const int sh = 2 * (7 - t);
    const int j0 = (rv0 >> sh) & 3;
    const int j1 = (rv1 >> sh) & 3;
    float2 m00 = lmsel(d0, j0, lA, lB, lC, lD, lE);
    float2 m10 = lmsel(d1, j0, lA, lB, lC, lD, lE);
    float2 m01 = lmsel(d0, j1, lA, lB, lC, lD, lE);
    float2 m11 = lmsel(d1, j1, lA, lB, lC, lD, lE);
    v2f br0 = {m00.x, m10.x}, bi0 = {m00.y, m10.y};
    v2f br1 = {m01.x, m11.x}, bi1 = {m01.y, m11.y};
    cr0 = __builtin_amdgcn_wmma_f32_16x16x4_f32(false, a, false, br0, (short)0,
                                                cr0, false, false);
    ci0 = __builtin_amdgcn_wmma_f32_16x16x4_f32(false, a, false, bi0, (short)0,
                                                ci0, false, false);
    cr1 = __builtin_amdgcn_wmma_f32_16x16x4_f32(false, a, false, br1, (short)0,
                                                cr1, false, false);
    ci1 = __builtin_amdgcn_wmma_f32_16x16x4_f32(false, a, false, bi1, (short)0,
                                                ci1, false, false);
  }

  // ---- epilogue: psi = (1/sqrt32) * sum_r exp(S[x][r]);  out = log(psi) ----
  // C layout: element v of lane l holds row (v + 8*half), col ln (per tile).
#pragma unroll
  for (int v = 0; v < 8; ++v) {
    float e0 = expf(cr0[v]);
    float zr = e0 * cosf(ci0[v]);
    float zi = e0 * sinf(ci0[v]);
    float e1 = expf(cr1[v]);
    zr += e1 * cosf(ci1[v]);
    zi += e1 * sinf(ci1[v]);
#pragma unroll
    for (int m = 1; m < 16; m <<= 1) {  // reduce cols within each 16-lane half
      zr += __shfl_xor(zr, m, 32);
      zi += __shfl_xor(zi, m, 32);
    }
    if (ln == 0) {  // lanes 0 and 16 own rows 0-7 and 8-15
      int row = half * 8 + v;
      float lr = 0.5f * logf(zr * zr + zi * zi) - 1.7328679513998633f; // -log(sqrt32)
      float li = atan2f(zi, zr);
      float2* o = (float2*)out;
      o[qbase + row] = make_float2(lr, li);
    }
  }
}

extern "C" void kernel_launch(void* const* d_in, const int* in_sizes, int n_in,
                              void* d_out, int out_size, void* d_ws,
                              size_t ws_size, hipStream_t stream) {
  const int* x_in = (const int*)d_in[0];
  const float* r1 = (const float*)d_in[1];
  const float* r2 = (const float*)d_in[2];
  const float* t1 = (const float*)d_in[3];
  const float* t2 = (const float*)d_in[4];
  float* out = (float*)d_out;  // complex64 -> interleaved (re, im) floats

  const int nq = in_sizes[0] / 24;   // 4096 query states
  const int blocks = nq / 16;        // one wave32 per 16 queries
  rvb_logpsi_kernel<<<blocks, 32, 0, stream>>>(x_in, r1, r2, t1, t2, out);
}